// Model_11879879543882
// MI455X (gfx1250) — compile-verified
//
#include <hip/hip_runtime.h>

#define KDIM 64
#define NDIM 64
#define NEXP 8

typedef __attribute__((ext_vector_type(2))) float v2f;
typedef __attribute__((ext_vector_type(8))) float v8f;

// d_ws int layout:
//   [0..7]                counts per expert
//   [8..15]               scatter cursors per expert
//   [16..24]              bucket bases (9 entries, 16-aligned slots; bases[8] = total)
//   [32 .. 32+permSlots)  perm[] (token index per slot, -1 = padding)
//   [32+permSlots .. +128) dump area for padding-row stores (garbage sink)

__global__ void cvmm_init(int* __restrict__ ws, int permSlots) {
    int tid = blockIdx.x * blockDim.x + threadIdx.x;
    if (tid < 32) ws[tid] = 0;
    if (tid < permSlots) ws[32 + tid] = -1;
}

__global__ void cvmm_hist(const int* __restrict__ sel, int M, int* __restrict__ ws) {
    __shared__ int lc[NEXP];
    int t = threadIdx.x;
    if (t < NEXP) lc[t] = 0;
    __syncthreads();
    int m = blockIdx.x * blockDim.x + t;
    if (m < M) atomicAdd(&lc[sel[m]], 1);
    __syncthreads();
    if (t < NEXP) atomicAdd(&ws[t], lc[t]);
}

__global__ void cvmm_scan(int* __restrict__ ws) {
    if (threadIdx.x == 0) {
        int base = 0;
        for (int e = 0; e < NEXP; ++e) {
            ws[16 + e] = base;
            base += (ws[e] + 15) & ~15;   // 16-align each bucket
        }
        ws[16 + NEXP] = base;
    }
}

__global__ void cvmm_scatter(const int* __restrict__ sel, int M, int* __restrict__ ws) {
    __shared__ int lc[NEXP];
    __shared__ int lbase[NEXP];
    int t = threadIdx.x;
    if (t < NEXP) lc[t] = 0;
    __syncthreads();
    int m = blockIdx.x * blockDim.x + t;
    int e = 0, lpos = 0;
    if (m < M) { e = sel[m]; lpos = atomicAdd(&lc[e], 1); }
    __syncthreads();
    if (t < NEXP) lbase[t] = atomicAdd(&ws[8 + t], lc[t]);
    __syncthreads();
    if (m < M) {
        int* perm = ws + 32;
        perm[ws[16 + e] + lbase[e] + lpos] = m;
    }
}

// One wave computes one 16-token x 64-col tile of the expert-sorted output.
// blockDim = 256 -> 8 waves -> 8 consecutive tiles per workgroup.
__global__ void __launch_bounds__(256)
cvmm_gemm(const float* __restrict__ x, const float* __restrict__ w,
          float* __restrict__ out, const int* __restrict__ ws,
          int numTiles, float* __restrict__ dump) {
    const int lane   = threadIdx.x & 31;
    const int wave   = threadIdx.x >> 5;
    int tile = blockIdx.x * 8 + wave;
    if (tile >= numTiles) tile = numTiles - 1;   // trailing waves redo last tile (same values)
    const int laneHi = lane >> 4;
    const int laneLo = lane & 15;

    const int* bases = ws + 16;
    const int* perm  = ws + 32;
    const int  slot0 = tile * 16;

    // Expert of this tile (buckets are 16-aligned, uniform per tile).
    int e = 0;
    #pragma unroll
    for (int i = 1; i < NEXP; ++i)
        if (slot0 >= bases[i]) e = i;

    // A fragments: 16x4 f32 layout — lanes 0-15 hold M=0..15 with K={0,1},
    // lanes 16-31 same M with K={2,3}. Padding rows gather x[0] (harmless;
    // their output is redirected to the dump and never reaches d_out).
    int rowTok = perm[slot0 + laneLo];
    if (rowTok < 0) rowTok = 0;
    const float* xrow = x + (size_t)rowTok * KDIM;

    v2f afrag[16];
    #pragma unroll
    for (int k = 0; k < 16; ++k)
        afrag[k] = __builtin_nontemporal_load((const v2f*)(xrow + k * 4 + laneHi * 2));

    // Branchless per-row destination: valid rows -> out row, padding -> dump.
    // D vgpr r holds M=r (lanes 0-15) / M=8+r (lanes 16-31), N = laneLo (+16n).
    float* rowptr[8];
    #pragma unroll
    for (int r = 0; r < 8; ++r) {
        int t = perm[slot0 + laneHi * 8 + r];
        rowptr[r] = (t >= 0) ? (out + (size_t)t * NDIM + laneLo) : (dump + lane);
    }

    const float* we = w + (size_t)e * (KDIM * NDIM);

    // k-outer / n-inner: 4 independent WMMA accumulation chains.
    v8f acc0 = (v8f){0,0,0,0,0,0,0,0};
    v8f acc1 = (v8f){0,0,0,0,0,0,0,0};
    v8f acc2 = (v8f){0,0,0,0,0,0,0,0};
    v8f acc3 = (v8f){0,0,0,0,0,0,0,0};

    #pragma unroll
    for (int k = 0; k < 16; ++k) {
        const int kb = k * 4 + laneHi * 2;
        const float* w0 = we + kb * NDIM + laneLo;
        const float* w1 = w0 + NDIM;
        v2f b0, b1, b2, b3;
        b0.x = w0[0];  b0.y = w1[0];
        b1.x = w0[16]; b1.y = w1[16];
        b2.x = w0[32]; b2.y = w1[32];
        b3.x = w0[48]; b3.y = w1[48];
        acc0 = __builtin_amdgcn_wmma_f32_16x16x4_f32(false, afrag[k], false, b0, (short)0, acc0, false, false);
        acc1 = __builtin_amdgcn_wmma_f32_16x16x4_f32(false, afrag[k], false, b1, (short)0, acc1, false, false);
        acc2 = __builtin_amdgcn_wmma_f32_16x16x4_f32(false, afrag[k], false, b2, (short)0, acc2, false, false);
        acc3 = __builtin_amdgcn_wmma_f32_16x16x4_f32(false, afrag[k], false, b3, (short)0, acc3, false, false);
    }

    // Unconditional non-temporal stores (write-once stream).
    #pragma unroll
    for (int r = 0; r < 8; ++r) {
        __builtin_nontemporal_store(acc0[r], rowptr[r] + 0);
        __builtin_nontemporal_store(acc1[r], rowptr[r] + 16);
        __builtin_nontemporal_store(acc2[r], rowptr[r] + 32);
        __builtin_nontemporal_store(acc3[r], rowptr[r] + 48);
    }
}

extern "C" void kernel_launch(void* const* d_in, const int* in_sizes, int n_in,
                              void* d_out, int out_size, void* d_ws, size_t ws_size,
                              hipStream_t stream) {
    const float* x   = (const float*)d_in[0];
    const int*   sel = (const int*)d_in[1];
    const float* w   = (const float*)d_in[2];
    float*       out = (float*)d_out;
    int*         ws  = (int*)d_ws;

    const int M = in_sizes[1];                 // token count
    const int permSlots = M + 16 * NEXP;       // worst-case padded slots
    const int numTiles  = permSlots / 16;
    float* dump = (float*)(ws + 32 + permSlots);

    cvmm_init   <<<(permSlots + 255) / 256, 256, 0, stream>>>(ws, permSlots);
    cvmm_hist   <<<(M + 255) / 256,        256, 0, stream>>>(sel, M, ws);
    cvmm_scan   <<<1,                       32, 0, stream>>>(ws);
    cvmm_scatter<<<(M + 255) / 256,        256, 0, stream>>>(sel, M, ws);
    cvmm_gemm   <<<(numTiles + 7) / 8,     256, 0, stream>>>(x, w, out, ws, numTiles, dump);
}